// TAGConv_5205500363145
// MI455X (gfx1250) — compile-verified
//
#include <hip/hip_runtime.h>

typedef float v2f __attribute__((ext_vector_type(2)));
typedef float v8f __attribute__((ext_vector_type(8)));

#define DIN  64
#define DCAT 192   // feat | hop1 | hop2

// ---------------------------------------------------------------------------
// init: Fcat[:,0:64] = feat, Fcat[:,64:192] = 0, deg = 0
// ---------------------------------------------------------------------------
__global__ void k_init(const float* __restrict__ feat, float* __restrict__ Fcat,
                       float* __restrict__ deg, int n) {
    int idx = blockIdx.x * blockDim.x + threadIdx.x;
    if (idx >= n * DIN) return;
    int node = idx >> 6;
    int d    = idx & 63;
    float* row = Fcat + (size_t)node * DCAT;
    row[d]        = feat[idx];
    row[64  + d]  = 0.0f;
    row[128 + d]  = 0.0f;
    if (d == 0) deg[node] = 0.0f;
}

// ---------------------------------------------------------------------------
// in-degree via fp32 global atomics (lowers to global_atomic_add_f32)
// ---------------------------------------------------------------------------
__global__ void k_degree(const long long* __restrict__ dst, float* __restrict__ deg, int e) {
    int i = blockIdx.x * blockDim.x + threadIdx.x;
    if (i >= e) return;
    unsafeAtomicAdd(&deg[(int)dst[i]], 1.0f);
}

__global__ void k_norm(const float* __restrict__ deg, float* __restrict__ norm, int n) {
    int i = blockIdx.x * blockDim.x + threadIdx.x;
    if (i >= n) return;
    norm[i] = rsqrtf(fmaxf(deg[i], 1.0f));
}

// ---------------------------------------------------------------------------
// scatter one hop: Fcat[dst, colbase:colbase+64] += h[src] * norm[src]
// 16 threads per edge; each moves 4 contiguous floats (float4 gather,
// 4x global_atomic_add_f32 scatter). Pre-scale by norm[src] fused in.
// ---------------------------------------------------------------------------
__global__ void k_scatter(const long long* __restrict__ src,
                          const long long* __restrict__ dst,
                          const float* __restrict__ h, int hstride,
                          const float* __restrict__ norm,
                          float* __restrict__ Fcat, int colbase, int e) {
    int t    = blockIdx.x * blockDim.x + threadIdx.x;
    int edge = t >> 4;
    if (edge >= e) return;
    int part = (t & 15) << 2;          // float offset within the 64-wide row

    int s = (int)src[edge];
    int d = (int)dst[edge];

    float4 v  = *(const float4*)(h + (size_t)s * hstride + part);
    float  ns = norm[s];

    float* drow = Fcat + (size_t)d * DCAT + colbase + part;
    unsafeAtomicAdd(drow + 0, v.x * ns);
    unsafeAtomicAdd(drow + 1, v.y * ns);
    unsafeAtomicAdd(drow + 2, v.z * ns);
    unsafeAtomicAdd(drow + 3, v.w * ns);
}

// post-scale the freshly accumulated block by norm[dst]
__global__ void k_postscale(float* __restrict__ Fcat, const float* __restrict__ norm,
                            int colbase, int n) {
    int idx = blockIdx.x * blockDim.x + threadIdx.x;
    if (idx >= n * DIN) return;
    int node = idx >> 6;
    int d    = idx & 63;
    Fcat[(size_t)node * DCAT + colbase + d] *= norm[node];
}

// ---------------------------------------------------------------------------
// out[n,o] = sum_k Fcat[n,k] * W[o,k] + bias[o]   via V_WMMA_F32_16X16X4_F32
// One wave -> 16 rows x 64 outputs (four 16x16 D tiles, K looped 192/4).
// W transposed into LDS as Wt[k][o] for conflict-free lane-consecutive reads.
// ---------------------------------------------------------------------------
__global__ void __launch_bounds__(256) k_gemm(const float* __restrict__ Fcat,
                                              const float* __restrict__ W,
                                              const float* __restrict__ bias,
                                              float* __restrict__ out, int n) {
    __shared__ float Wt[DCAT * 64];        // 48 KB: Wt[k*64 + o] = W[o*192 + k]
    for (int i = threadIdx.x; i < DCAT * 64; i += blockDim.x) {
        int o = i / DCAT;
        int k = i - o * DCAT;
        Wt[k * 64 + o] = W[i];
    }
    __syncthreads();

    int wave = threadIdx.x >> 5;
    int lane = threadIdx.x & 31;
    int row0 = (blockIdx.x * 8 + wave) * 16;
    if (row0 >= n) return;                 // whole wave uniform -> EXEC all-1 at WMMA

    // A-matrix layout (32-bit 16x4): lanes 0-15 rows M=0..15 hold K=0,1;
    // lanes 16-31 same rows hold K=2,3.
    int m  = lane & 15;                    // A row in tile == B/D column N
    int kh = (lane >> 4) << 1;             // 0 or 2
    int ar = min(row0 + m, n - 1);         // clamp keeps EXEC uniform
    const float* arow = Fcat + (size_t)ar * DCAT + kh;

    v8f acc0 = {}, acc1 = {}, acc2 = {}, acc3 = {};

#pragma unroll 4
    for (int k0 = 0; k0 < DCAT; k0 += 4) {
        v2f a;
        a.x = arow[k0];
        a.y = arow[k0 + 1];

        const float* bp = &Wt[(k0 + kh) * 64 + m];   // B: N=lane&15, K=kh(+1)
        v2f b0, b1, b2, b3;
        b0.x = bp[0];   b0.y = bp[64];               // N-tile 0  (o =  0..15)
        b1.x = bp[16];  b1.y = bp[80];               // N-tile 1  (o = 16..31)
        b2.x = bp[32];  b2.y = bp[96];               // N-tile 2  (o = 32..47)
        b3.x = bp[48];  b3.y = bp[112];              // N-tile 3  (o = 48..63)

        acc0 = __builtin_amdgcn_wmma_f32_16x16x4_f32(false, a, false, b0, (short)0, acc0, false, false);
        acc1 = __builtin_amdgcn_wmma_f32_16x16x4_f32(false, a, false, b1, (short)0, acc1, false, false);
        acc2 = __builtin_amdgcn_wmma_f32_16x16x4_f32(false, a, false, b2, (short)0, acc2, false, false);
        acc3 = __builtin_amdgcn_wmma_f32_16x16x4_f32(false, a, false, b3, (short)0, acc3, false, false);
    }

    // D layout: VGPR j -> M = j (lanes 0-15) or j+8 (lanes 16-31); N = lane&15
    int   half = (lane >> 4) << 3;
    float bs0 = bias[m], bs1 = bias[16 + m], bs2 = bias[32 + m], bs3 = bias[48 + m];

#pragma unroll
    for (int j = 0; j < 8; ++j) {
        int r = row0 + half + j;
        if (r >= n) continue;
        float* orow = out + (size_t)r * 64 + m;
        orow[0]  = acc0[j] + bs0;
        orow[16] = acc1[j] + bs1;
        orow[32] = acc2[j] + bs2;
        orow[48] = acc3[j] + bs3;
    }
}

// ---------------------------------------------------------------------------
extern "C" void kernel_launch(void* const* d_in, const int* in_sizes, int n_in,
                              void* d_out, int out_size, void* d_ws, size_t ws_size,
                              hipStream_t stream) {
    const float*     feat = (const float*)d_in[0];
    const float*     W    = (const float*)d_in[1];   // [64, 192] row-major
    const float*     bias = (const float*)d_in[2];
    const long long* src  = (const long long*)d_in[3]; // reference dtype int64
    const long long* dst  = (const long long*)d_in[4];

    int n = in_sizes[0] / DIN;   // 50000
    int e = in_sizes[3];         // 1600000

    // workspace layout (floats): deg[n] | norm[n] | Fcat[n*192]
    float* ws   = (float*)d_ws;
    float* deg  = ws;
    float* norm = ws + n;
    float* Fcat = ws + 2 * (size_t)n;

    int tpb = 256;
    int gN  = (n * DIN + tpb - 1) / tpb;
    int gE  = (e + tpb - 1) / tpb;
    int gE16 = (int)(((size_t)e * 16 + tpb - 1) / tpb);

    k_init  <<<gN, tpb, 0, stream>>>(feat, Fcat, deg, n);
    k_degree<<<gE, tpb, 0, stream>>>(dst, deg, e);
    k_norm  <<<(n + tpb - 1) / tpb, tpb, 0, stream>>>(deg, norm, n);

    // hop 1: feat (stride 64) -> Fcat cols 64..127
    k_scatter  <<<gE16, tpb, 0, stream>>>(src, dst, feat, DIN, norm, Fcat, 64, e);
    k_postscale<<<gN,   tpb, 0, stream>>>(Fcat, norm, 64, n);

    // hop 2: Fcat cols 64..127 (stride 192) -> Fcat cols 128..191
    k_scatter  <<<gE16, tpb, 0, stream>>>(src, dst, Fcat + 64, DCAT, norm, Fcat, 128, e);
    k_postscale<<<gN,   tpb, 0, stream>>>(Fcat, norm, 128, n);

    // projection: [n,192] @ [192,64] + bias, WMMA f32 16x16x4
    int rowTiles = (n + 15) / 16;
    int gG = (rowTiles + 7) / 8;        // 8 waves per block, 16 rows per wave
    k_gemm<<<gG, tpb, 0, stream>>>(Fcat, W, bias, (float*)d_out, n);
}